// DotProductAttention_25331717112244
// MI455X (gfx1250) — compile-verified
//
#include <hip/hip_runtime.h>
#include <stdint.h>

typedef __attribute__((ext_vector_type(2))) float v2f;
typedef __attribute__((ext_vector_type(4))) float v4f;
typedef __attribute__((ext_vector_type(8))) float v8f;
typedef int gv4i __attribute__((vector_size(16)));   // matches builtin's int4 pointee

#define B_  32
#define Q_  1024
#define K_  1024
#define D_  128
#define LDS_STRIDE 132            // 128 + 4 pad dwords -> conflict-friendly frag loads
#define TILE_F (16 * LDS_STRIDE)  // one 16x128 tile (padded), in floats

// ---- feature probes (compile-only environment: fall back cleanly) ----------
#if defined(__has_builtin)
#  if __has_builtin(__builtin_amdgcn_global_load_async_to_lds_b128)
#    define HAS_ASYNC_LDS 1
#  endif
#  if __has_builtin(__builtin_amdgcn_ds_swizzle)
#    define HAS_SWIZZLE 1
#  endif
#endif

#if defined(HAS_ASYNC_LDS)
#  if __has_builtin(__builtin_amdgcn_s_wait_asynccnt)
#    define WAIT_ASYNC() __builtin_amdgcn_s_wait_asynccnt(0)
#  else
#    define WAIT_ASYNC() asm volatile("s_wait_asynccnt 0x0" ::: "memory")
#  endif
#else
#  define WAIT_ASYNC() do {} while (0)
#endif

__device__ __forceinline__ float swap_half(float x) {
  // exchange lanes l <-> l+16 (wave32): ds_swizzle SWAPX16 (xor=0x10,and=0x1f)
#if defined(HAS_SWIZZLE) && defined(__AMDGCN__)
  return __int_as_float(__builtin_amdgcn_ds_swizzle(__float_as_int(x), 0x401f));
#else
  return __shfl_xor(x, 16, 32);
#endif
}

// Cooperative copy of one K tile + one V tile (16 rows x 128 f32 each) into LDS.
// 256 threads x 4 chunks of float4 = 1024 chunks. Uses gfx1250 async DMA when present.
__device__ __forceinline__ void copy_tiles(const float* __restrict__ gK,
                                           const float* __restrict__ gV,
                                           float* lds, int tid, int kb) {
#pragma unroll
  for (int j = 0; j < 4; ++j) {
    int c    = tid + 256 * j;
    int tile = c >> 9;           // 0 = K, 1 = V
    int row  = (c >> 5) & 15;    // key row within tile
    int col  = (c & 31) * 4;     // float4 column
    const float* g = (tile ? gV : gK) + (size_t)(kb * 16 + row) * D_ + col;
    float*       l = lds + tile * TILE_F + row * LDS_STRIDE + col;
#if defined(HAS_ASYNC_LDS)
    __builtin_amdgcn_global_load_async_to_lds_b128(
        (__attribute__((address_space(1))) gv4i*)(uintptr_t)g,
        (__attribute__((address_space(3))) gv4i*)(uint32_t)(uintptr_t)l,
        0, 0);
#else
    *(v4f*)l = *(const v4f*)g;
#endif
  }
}

__global__ __launch_bounds__(256, 1)
void fa_wmma_f32_kernel(const float* __restrict__ Qm, const float* __restrict__ Km,
                        const float* __restrict__ Vm, const int* __restrict__ VL,
                        float* __restrict__ Om) {
  __shared__ float smem[2 * 2 * TILE_F];   // double-buffered {K,V} tiles

  const int tid  = threadIdx.x;
  const int b    = blockIdx.x >> 3;        // 8 q-chunks of 128 per batch
  const int qc   = blockIdx.x & 7;
  const int wave = tid >> 5;
  const int lane = tid & 31;
  const int ln   = lane & 15;              // row/col-within-halfwave
  const int hh   = lane >> 4;              // halfwave id
  const bool hi  = (hh != 0);
  const int q0   = qc * 128 + wave * 16;   // this wave's 16 queries

  const int vlen = VL[b];
  const int nkb  = (vlen + 15) >> 4;       // skip fully-masked key blocks (exact)
  const float scale = 0.08838834764831845f; // 1/sqrt(128)

  const float* gK = Km + (size_t)b * K_ * D_;
  const float* gV = Vm + (size_t)b * K_ * D_;

  // ---- Q fragments, pre-scaled (B-frag of Q^T: lane=q, K-dim split by halfwave)
  v2f qf[32];
  {
    const float* qrow = Qm + ((size_t)b * Q_ + q0 + ln) * D_ + 2 * hh;
#pragma unroll
    for (int s = 0; s < 32; ++s) {
      v2f t = *(const v2f*)(qrow + 4 * s);
      qf[s].x = t.x * scale;
      qf[s].y = t.y * scale;
    }
  }

  v8f acc[8];
#pragma unroll
  for (int t = 0; t < 8; ++t) {
    v8f z = {0.f, 0.f, 0.f, 0.f, 0.f, 0.f, 0.f, 0.f};
    acc[t] = z;
  }
  float m    = -__builtin_inff();
  float lsum = 0.0f;

  // ---- prologue: stage first K/V tile
  copy_tiles(gK, gV, smem, tid, 0);
  WAIT_ASYNC();
  __syncthreads();

  for (int kb = 0; kb < nkb; ++kb) {
    const int cur = kb & 1;
    if (kb + 1 < nkb)                       // prefetch next tile into other buffer
      copy_tiles(gK, gV, smem + (cur ^ 1) * 2 * TILE_F, tid, kb + 1);

    const float* ldsK = smem + cur * 2 * TILE_F;
    const float* ldsV = ldsK + TILE_F;

    // ---- S^T(16k x 16q) = K_tile(16x128) x Q^T(128x16)
    // Software-pipelined fragment preload (8 frags/chunk, double buffered) and
    // dual accumulators to break the 32-deep WMMA RAW chain.
    const float* kfp = ldsK + ln * LDS_STRIDE + 2 * hh;
    v8f s8a = {0.f, 0.f, 0.f, 0.f, 0.f, 0.f, 0.f, 0.f};
    v8f s8b = {0.f, 0.f, 0.f, 0.f, 0.f, 0.f, 0.f, 0.f};
    v2f kfrag[2][8];
#pragma unroll
    for (int s = 0; s < 8; ++s) kfrag[0][s] = *(const v2f*)(kfp + 4 * s);
#pragma unroll
    for (int c = 0; c < 4; ++c) {
      const int cb = c & 1;
      if (c < 3) {
#pragma unroll
        for (int s = 0; s < 8; ++s)
          kfrag[cb ^ 1][s] = *(const v2f*)(kfp + 4 * (8 * (c + 1) + s));
      }
#pragma unroll
      for (int s = 0; s < 8; s += 2) {
        s8a = __builtin_amdgcn_wmma_f32_16x16x4_f32(false, kfrag[cb][s], false,
                                                    qf[8 * c + s], (short)0,
                                                    s8a, false, false);
        s8b = __builtin_amdgcn_wmma_f32_16x16x4_f32(false, kfrag[cb][s + 1], false,
                                                    qf[8 * c + s + 1], (short)0,
                                                    s8b, false, false);
      }
    }

    // ---- mask + online softmax (keys live in the VGPR dim; lane-halves split keys)
    const int key0 = kb * 16 + 8 * hh;
    float sc[8];
    float mloc = -__builtin_inff();
#pragma unroll
    for (int r = 0; r < 8; ++r) {
      float x = s8a[r] + s8b[r];
      x = (key0 + r < vlen) ? x : -1e9f;    // same masking constant as reference
      sc[r] = x;
      mloc  = fmaxf(mloc, x);
    }
    float mblk  = fmaxf(mloc, swap_half(mloc));
    float mnew  = fmaxf(m, mblk);
    float alpha = __expf(m - mnew);         // first iter: exp(-inf)=0, no NaN

    float p[8];
    float ps = 0.f;
#pragma unroll
    for (int r = 0; r < 8; ++r) {
      p[r] = __expf(sc[r] - mnew);
      ps  += p[r];
    }
    lsum = lsum * alpha + ps + swap_half(ps);
    m    = mnew;

#pragma unroll
    for (int t = 0; t < 8; ++t)
#pragma unroll
      for (int r = 0; r < 8; ++r) acc[t][r] *= alpha;

    // ---- P^T D-layout -> B-fragments: 8 half-wave swaps + selects
    float sw[8];
#pragma unroll
    for (int r = 0; r < 8; ++r) sw[r] = swap_half(p[r]);
    v2f pb[4];
    pb[0].x = hi ? sw[2] : p[0];  pb[0].y = hi ? sw[3] : p[1];  // keys 0..3
    pb[1].x = hi ? sw[6] : p[4];  pb[1].y = hi ? sw[7] : p[5];  // keys 4..7
    pb[2].x = hi ? p[2] : sw[0];  pb[2].y = hi ? p[3] : sw[1];  // keys 8..11
    pb[3].x = hi ? p[6] : sw[4];  pb[3].y = hi ? p[7] : sw[5];  // keys 12..15

    // ---- O^T(128d x 16q) += V^T x P^T : 8 d-tiles x 4 k-steps
    // Preload all 8 V-fragments of a k-step, then fire 8 independent WMMAs.
    const float* vfp = ldsV + (2 * hh) * LDS_STRIDE + ln;
#pragma unroll
    for (int j = 0; j < 4; ++j) {
      v2f vfj[8];
#pragma unroll
      for (int t = 0; t < 8; ++t) {
        vfj[t].x = vfp[(4 * j) * LDS_STRIDE + 16 * t];
        vfj[t].y = vfp[(4 * j + 1) * LDS_STRIDE + 16 * t];
      }
#pragma unroll
      for (int t = 0; t < 8; ++t)
        acc[t] = __builtin_amdgcn_wmma_f32_16x16x4_f32(false, vfj[t], false, pb[j],
                                                       (short)0, acc[t], false, false);
    }

    WAIT_ASYNC();
    __syncthreads();
  }

  // ---- normalize + store: lane holds q=q0+ln, 8 contiguous d per tile
  const float inv = 1.0f / lsum;
  float* orow = Om + ((size_t)b * Q_ + q0 + ln) * D_ + 8 * hh;
#pragma unroll
  for (int t = 0; t < 8; ++t) {
    v4f o0, o1;
    o0.x = acc[t][0] * inv;  o0.y = acc[t][1] * inv;
    o0.z = acc[t][2] * inv;  o0.w = acc[t][3] * inv;
    o1.x = acc[t][4] * inv;  o1.y = acc[t][5] * inv;
    o1.z = acc[t][6] * inv;  o1.w = acc[t][7] * inv;
    *(v4f*)(orow + 16 * t)     = o0;
    *(v4f*)(orow + 16 * t + 4) = o1;
  }
}

extern "C" void kernel_launch(void* const* d_in, const int* in_sizes, int n_in,
                              void* d_out, int out_size, void* d_ws, size_t ws_size,
                              hipStream_t stream) {
  const float* q  = (const float*)d_in[0];
  const float* k  = (const float*)d_in[1];
  const float* v  = (const float*)d_in[2];
  const int*   vl = (const int*)d_in[3];
  float* out = (float*)d_out;
  (void)in_sizes; (void)n_in; (void)out_size; (void)d_ws; (void)ws_size;

  dim3 grid(B_ * (Q_ / 128));   // 256 workgroups: one batch x 128-query chunk each
  dim3 block(256);              // 8 waves (wave32) -> 8 q-tiles of 16
  fa_wmma_f32_kernel<<<grid, block, 0, stream>>>(q, k, v, vl, out);
}